// RigidBody_24215025615351
// MI455X (gfx1250) — compile-verified
//
#include <hip/hip_runtime.h>
#include <hip/hip_bf16.h>

typedef __attribute__((ext_vector_type(2))) float v2f;
typedef __attribute__((ext_vector_type(8))) float v8f;

#define NBODY 16
#define NBATCH 4096
#define NQ 32
#define KPAIRS 64      // 4 K-pairs per body: {Js01},{Js2,0},{Jr01},{Jr2,0} -> K=128
#define CS 48          // float2 entries per kpair row (cols 0..47; 32 data + f + g + zero pad)

__global__ __launch_bounds__(32)
void rigid_crba_wmma(const float* __restrict__ Js,   // [L,B,3,32]
                     const float* __restrict__ Jr,   // [L,B,3,32]
                     const float* __restrict__ Rib,  // [L,B,3,3]
                     const float* __restrict__ Wv,   // [L,B,3]
                     const float* __restrict__ dWv,  // [L,B,3]
                     const float* __restrict__ Av,   // [L,B,3]
                     const float* __restrict__ Mm,   // [L]
                     const float* __restrict__ Ib,   // [L,3,3]
                     const float* __restrict__ Gv,   // [3]
                     float* __restrict__ out)        // [B,32,34]
{
    extern __shared__ float lds[];
    v2f* A2 = (v2f*)lds;              // KPAIRS x CS  (float2 per entry)
    v2f* B2 = A2 + KPAIRS * CS;       // KPAIRS x CS

    const int b    = blockIdx.x;
    const int lane = threadIdx.x;     // 0..31, one full wave32
    const int half = lane >> 4;       // 0: lanes 0-15, 1: lanes 16-31
    const int lcol = lane & 15;

    const float g0 = Gv[0], g1 = Gv[1], g2 = Gv[2];

    // ---- Phase 1: build fragment-native A (K x 32) and Bext (K x 34) in LDS ----
    for (int l = 0; l < NBODY; ++l) {
        const size_t jbase = ((size_t)(l * NBATCH + b)) * 96;   // 3*32 floats
        // coalesced per-lane row loads (col = lane)
        float js0 = Js[jbase + 0 * 32 + lane];
        float js1 = Js[jbase + 1 * 32 + lane];
        float js2 = Js[jbase + 2 * 32 + lane];
        float jr0 = Jr[jbase + 0 * 32 + lane];
        float jr1 = Jr[jbase + 1 * 32 + lane];
        float jr2 = Jr[jbase + 2 * 32 + lane];
        if (l + 1 < NBODY) {
            __builtin_prefetch(Js + jbase + (size_t)NBATCH * 96, 0, 1);
            __builtin_prefetch(Jr + jbase + (size_t)NBATCH * 96, 0, 1);
        }
        // block-uniform scalars -> SMEM
        const float ml = Mm[l];
        const float* Il = Ib + l * 9;
        float i00 = Il[0], i01 = Il[1], i02 = Il[2];
        float i10 = Il[3], i11 = Il[4], i12 = Il[5];
        float i20 = Il[6], i21 = Il[7], i22 = Il[8];
        const size_t vb = ((size_t)(l * NBATCH + b)) * 3;
        float w0 = Wv[vb], w1 = Wv[vb + 1], w2 = Wv[vb + 2];
        float d0 = dWv[vb], d1 = dWv[vb + 1], d2 = dWv[vb + 2];
        float a0 = Av[vb], a1 = Av[vb + 1], a2 = Av[vb + 2];
        const size_t rb = ((size_t)(l * NBATCH + b)) * 9;
        // gB[k] = sum_j Rib[j][k] * grav[j]   (A_IB^T grav)
        float gb0 = Rib[rb + 0] * g0 + Rib[rb + 3] * g1 + Rib[rb + 6] * g2;
        float gb1 = Rib[rb + 1] * g0 + Rib[rb + 4] * g1 + Rib[rb + 7] * g2;
        float gb2 = Rib[rb + 2] * g0 + Rib[rb + 5] * g1 + Rib[rb + 8] * g2;
        // rot = I*dw + w x (I*w)
        float iw0 = i00 * w0 + i01 * w1 + i02 * w2;
        float iw1 = i10 * w0 + i11 * w1 + i12 * w2;
        float iw2 = i20 * w0 + i21 * w1 + i22 * w2;
        float r0 = i00 * d0 + i01 * d1 + i02 * d2 + (w1 * iw2 - w2 * iw1);
        float r1 = i10 * d0 + i11 * d1 + i12 * d2 + (w2 * iw0 - w0 * iw2);
        float r2 = i20 * d0 + i21 * d1 + i22 * d2 + (w0 * iw1 - w1 * iw0);
        // I*Jr rows
        float q0 = i00 * jr0 + i01 * jr1 + i02 * jr2;
        float q1 = i10 * jr0 + i11 * jr1 + i12 * jr2;
        float q2 = i20 * jr0 + i21 * jr1 + i22 * jr2;

        const int kp = l * 4;
        // A kpairs: {m*Js0, m*Js1}, {m*Js2, 0}, {Jr0, Jr1}, {Jr2, 0}
        A2[(kp + 0) * CS + lane] = (v2f){ ml * js0, ml * js1 };
        A2[(kp + 1) * CS + lane] = (v2f){ ml * js2, 0.0f };
        A2[(kp + 2) * CS + lane] = (v2f){ jr0, jr1 };
        A2[(kp + 3) * CS + lane] = (v2f){ jr2, 0.0f };
        // B kpairs: {Js0, Js1}, {Js2, 0}, {IJr0, IJr1}, {IJr2, 0}
        B2[(kp + 0) * CS + lane] = (v2f){ js0, js1 };
        B2[(kp + 1) * CS + lane] = (v2f){ js2, 0.0f };
        B2[(kp + 2) * CS + lane] = (v2f){ q0, q1 };
        B2[(kp + 3) * CS + lane] = (v2f){ q2, 0.0f };
        // B columns 32..47: col32 = f-coeff, col33 = g-coeff, rest zero
        if (lane < 16) {
            const int c = 32 + lane;
            v2f fv[4] = { { -a0, -a1 }, { -a2, 0.0f }, { -r0, -r1 }, { -r2, 0.0f } };
            v2f gv[4] = { { gb0, gb1 }, { gb2, 0.0f }, { 0.0f, 0.0f }, { 0.0f, 0.0f } };
#pragma unroll
            for (int p = 0; p < 4; ++p) {
                v2f v = (lane == 0) ? fv[p] : (lane == 1 ? gv[p] : (v2f){ 0.0f, 0.0f });
                B2[(kp + p) * CS + c] = v;
            }
        }
    }

    __syncthreads();   // single-wave WG: orders DS writes before DS reads

    // ---- Phase 2: out(32x34) = A^T(32xK) * Bext(Kx34), K = 128 --------------
    v8f c00 = {}, c01 = {}, c02 = {}, c10 = {}, c11 = {}, c12 = {};
    for (int kp = 0; kp < KPAIRS; kp += 2) {
        // A 16x4 frag: lanes 0-15 take kpair kp (K=2kp..+1), lanes 16-31 kpair kp+1
        const int r = (kp + half) * CS;
        v2f a0 = A2[r + lcol];
        v2f a1 = A2[r + 16 + lcol];
        v2f b0 = B2[r + lcol];
        v2f b1 = B2[r + 16 + lcol];
        v2f b2 = B2[r + 32 + lcol];
        c00 = __builtin_amdgcn_wmma_f32_16x16x4_f32(false, a0, false, b0, (short)0, c00, false, false);
        c01 = __builtin_amdgcn_wmma_f32_16x16x4_f32(false, a0, false, b1, (short)0, c01, false, false);
        c02 = __builtin_amdgcn_wmma_f32_16x16x4_f32(false, a0, false, b2, (short)0, c02, false, false);
        c10 = __builtin_amdgcn_wmma_f32_16x16x4_f32(false, a1, false, b0, (short)0, c10, false, false);
        c11 = __builtin_amdgcn_wmma_f32_16x16x4_f32(false, a1, false, b1, (short)0, c11, false, false);
        c12 = __builtin_amdgcn_wmma_f32_16x16x4_f32(false, a1, false, b2, (short)0, c12, false, false);
    }

    // ---- Phase 3: store packed [32, 34] --------------------------------------
    float* ob = out + (size_t)b * 32 * 34;
#pragma unroll
    for (int v = 0; v < 8; ++v) {
        const int n0 = v + half * 8;          // C/D layout: VGPR v -> M = v (+8 for hi lanes)
        ob[n0 * 34 + lcol]             = c00[v];
        ob[n0 * 34 + 16 + lcol]        = c01[v];
        ob[(n0 + 16) * 34 + lcol]      = c10[v];
        ob[(n0 + 16) * 34 + 16 + lcol] = c11[v];
        if (lcol < 2) {                       // cols 32 (f), 33 (g)
            ob[n0 * 34 + 32 + lcol]        = c02[v];
            ob[(n0 + 16) * 34 + 32 + lcol] = c12[v];
        }
    }
}

extern "C" void kernel_launch(void* const* d_in, const int* in_sizes, int n_in,
                              void* d_out, int out_size, void* d_ws, size_t ws_size,
                              hipStream_t stream) {
    const float* Js  = (const float*)d_in[0];
    const float* Jr  = (const float*)d_in[1];
    const float* Rib = (const float*)d_in[2];
    const float* Wv  = (const float*)d_in[3];
    const float* dWv = (const float*)d_in[4];
    const float* Av  = (const float*)d_in[5];
    const float* Mm  = (const float*)d_in[6];
    const float* Ib  = (const float*)d_in[7];
    const float* Gv  = (const float*)d_in[8];
    float* out = (float*)d_out;
    (void)in_sizes; (void)n_in; (void)out_size; (void)d_ws; (void)ws_size;

    const size_t ldsBytes = (size_t)2 * KPAIRS * CS * sizeof(v2f);   // 48 KB
    rigid_crba_wmma<<<NBATCH, 32, ldsBytes, stream>>>(
        Js, Jr, Rib, Wv, dWv, Av, Mm, Ib, Gv, out);
}